// FSCILGate_30554397343879
// MI455X (gfx1250) — compile-verified
//
#include <hip/hip_runtime.h>
#include <hip/hip_bf16.h>

// ---------------------------------------------------------------------------
// MoE gating for MI455X (gfx1250, wave32):
//   logits = x @ q^T / T   via V_WMMA_F32_16X16X4_F32 (native fp32 WMMA)
//   gate   = softmax(logits)  (16 experts per row, shfl-based)
//   aux    = 0.01 * mean_e(avg_gate[e] * load[e]) * 256   (deterministic 2-stage)
// Memory-bound problem (~7.8 flop/B) -> fp32 WMMA + streamed x + q in LDS.
// ---------------------------------------------------------------------------

typedef __attribute__((ext_vector_type(2))) float v2f;
typedef __attribute__((ext_vector_type(8))) float v8f;

#define DDIM 512
#define NEXP 16
#define KSTEPS (DDIM / 4) // 128 wmma k-steps of K=4

// ---------------------------------------------------------------------------
// Kernel 1: GEMM + softmax. One wave = one 16-token tile.
//   A frag (16x4 f32): lane<16 holds row M=lane, K=0..1; lane>=16 holds K=2..3.
//   B frag (4x16 f32): lane = 16*half + n holds B[2*half + j][n], j = vgpr 0/1.
//   C (16x16 f32, 8 VGPRs): c[r] @ lane l -> row (r + 8*(l>>4)), col (l&15).
// q staged in LDS as float2 pairs keyed by (d/2, expert): bank-conflict free.
// ---------------------------------------------------------------------------
__global__ __launch_bounds__(256) void moe_gate_main(
    const float* __restrict__ x, const float* __restrict__ q,
    const float* __restrict__ temp, float* __restrict__ out, int num_tiles)
{
  __shared__ float qs[NEXP * DDIM]; // 32 KB

  const int tid = threadIdx.x;

  // Cooperative transposed load of expert_queries: pair p=d/2, expert e.
  {
    const float2* q2 = (const float2*)q;
    float2* s2 = (float2*)qs;
#pragma unroll
    for (int j = 0; j < 16; ++j) {
      int i = tid + 256 * j;   // flat float2 index over q (never straddles rows)
      float2 v = q2[i];
      int e = i >> 8;          // i / 256  (256 pairs per 512-elem row)
      int p = i & 255;         // d / 2
      s2[p * NEXP + e] = v;
    }
  }
  __syncthreads();

  const int wave = tid >> 5;
  const int lane = tid & 31;
  const int hf   = lane >> 4;  // half-wave id
  const int l16  = lane & 15;

  const int tile = blockIdx.x * 8 + wave;
  if (tile >= num_tiles) return;
  const int tok0 = tile * 16;

  const float*  xrow = x + (size_t)(tok0 + l16) * DDIM + 2 * hf;
  const float2* qp   = (const float2*)qs;

  v8f c0 = {};
  v8f c1 = {};
#pragma unroll 4
  for (int k = 0; k < KSTEPS; k += 2) {
    if ((k & 7) == 0)
      __builtin_prefetch(xrow + 4 * k + 128, 0, 3); // global_prefetch_b8, 512B ahead
    float2 af0 = *(const float2*)(xrow + 4 * k);
    float2 bf0 = qp[(2 * k + hf) * NEXP + l16];
    float2 af1 = *(const float2*)(xrow + 4 * (k + 1));
    float2 bf1 = qp[(2 * (k + 1) + hf) * NEXP + l16];
    v2f a0; a0.x = af0.x; a0.y = af0.y;
    v2f b0; b0.x = bf0.x; b0.y = bf0.y;
    v2f a1; a1.x = af1.x; a1.y = af1.y;
    v2f b1; b1.x = bf1.x; b1.y = bf1.y;
    c0 = __builtin_amdgcn_wmma_f32_16x16x4_f32(false, a0, false, b0, (short)0, c0, false, false);
    c1 = __builtin_amdgcn_wmma_f32_16x16x4_f32(false, a1, false, b1, (short)0, c1, false, false);
  }

  const float invT = 1.0f / temp[0];
  float gates[8];
#pragma unroll
  for (int r = 0; r < 8; ++r) {
    float lg = (c0[r] + c1[r]) * invT;
    // row of 16 experts lives in 16 lanes of one half-wave: xor masks 1..8 stay inside it
    float m = lg;
#pragma unroll
    for (int s = 1; s < 16; s <<= 1) m = fmaxf(m, __shfl_xor(m, s, 32));
    float ev = __expf(lg - m);
    float sum = ev;
#pragma unroll
    for (int s = 1; s < 16; s <<= 1) sum += __shfl_xor(sum, s, 32);
    gates[r] = ev / sum;
  }
#pragma unroll
  for (int r = 0; r < 8; ++r)
    out[(size_t)(tok0 + r + 8 * hf) * NEXP + l16] = gates[r];
}

// ---------------------------------------------------------------------------
// Kernel 2 (stage A): deterministic per-block partial sums of
//   gate_sum[e] and top-2 load_count[e] over 4096 tokens per block.
// ---------------------------------------------------------------------------
__global__ __launch_bounds__(256) void moe_gate_reduceA(
    const float* __restrict__ gates, float* __restrict__ partial, int ntok)
{
  __shared__ float sred[32][256]; // 32 KB
  const int tid = threadIdx.x;

  float accg[NEXP];
  float accl[NEXP];
#pragma unroll
  for (int e = 0; e < NEXP; ++e) { accg[e] = 0.f; accl[e] = 0.f; }

  const int base = blockIdx.x * 4096 + tid;
  for (int j = 0; j < 16; ++j) {
    int tok = base + j * 256;
    if (tok < ntok) {
      const float* g = gates + (size_t)tok * NEXP;
      float m1 = -3.4e38f, m2 = -3.4e38f;
      int i1 = 0, i2 = 0;
#pragma unroll
      for (int e = 0; e < NEXP; ++e) {
        float v = g[e];
        accg[e] += v;
        if (v > m1)      { m2 = m1; i2 = i1; m1 = v; i1 = e; } // strict > keeps lowest idx on tie
        else if (v > m2) { m2 = v;  i2 = e; }
      }
      accl[i1] += 1.0f;
      accl[i2] += 1.0f;
    }
  }

#pragma unroll
  for (int e = 0; e < NEXP; ++e) {
    sred[e][tid]        = accg[e];
    sred[NEXP + e][tid] = accl[e];
  }
  __syncthreads();
  for (int s = 128; s > 0; s >>= 1) {
    if (tid < s) {
#pragma unroll
      for (int k = 0; k < 32; ++k) sred[k][tid] += sred[k][tid + s];
    }
    __syncthreads();
  }
  if (tid < 32) partial[blockIdx.x * 32 + tid] = sred[tid][0];
}

// ---------------------------------------------------------------------------
// Kernel 3 (stage B): single-thread deterministic combine -> aux scalar.
// ---------------------------------------------------------------------------
__global__ void moe_gate_aux(const float* __restrict__ partial,
                             float* __restrict__ aux_out, int nblk, int ntok)
{
  if (threadIdx.x == 0 && blockIdx.x == 0) {
    float sg[NEXP], sl[NEXP];
    for (int e = 0; e < NEXP; ++e) { sg[e] = 0.f; sl[e] = 0.f; }
    for (int b = 0; b < nblk; ++b) {
      for (int e = 0; e < NEXP; ++e) {
        sg[e] += partial[b * 32 + e];
        sl[e] += partial[b * 32 + NEXP + e];
      }
    }
    const float fN = (float)ntok;
    float acc = 0.f;
    for (int e = 0; e < NEXP; ++e)
      acc += (sg[e] / fN) * (sl[e] / (2.0f * fN));
    aux_out[0] = 0.01f * (acc / (float)NEXP) * (float)(NEXP * NEXP);
  }
}

extern "C" void kernel_launch(void* const* d_in, const int* in_sizes, int n_in,
                              void* d_out, int out_size, void* d_ws, size_t ws_size,
                              hipStream_t stream)
{
  const float* x = (const float*)d_in[0];          // [B,H,W,D] f32
  const float* q = (const float*)d_in[1];          // [16,512]  f32
  const float* t = (const float*)d_in[2];          // [1]       f32
  float* out = (float*)d_out;                      // [N*16] gates ++ [1] aux

  const int ntok      = in_sizes[0] / DDIM;        // 200704
  const int num_tiles = ntok / 16;                 // 12544 (N divisible by 16)
  const int blocks    = (num_tiles + 7) / 8;       // 8 waves/block

  moe_gate_main<<<blocks, 256, 0, stream>>>(x, q, t, out, num_tiles);

  float* partial = (float*)d_ws;                   // nblkA*32 floats (~6 KB)
  const int nblkA = (ntok + 4095) / 4096;          // 49
  moe_gate_reduceA<<<nblkA, 256, 0, stream>>>(out, partial, ntok);
  moe_gate_aux<<<1, 32, 0, stream>>>(partial, out + (size_t)ntok * NEXP, nblkA, ntok);
}